// EquivariantProductBasisBlock_72335839199639
// MI455X (gfx1250) — compile-verified
//
#include <hip/hip_runtime.h>

typedef __attribute__((ext_vector_type(2))) float v2f;
typedef __attribute__((ext_vector_type(8))) float v8f;

#define NNODES 10000
#define CH     128
#define LSQ    9      // (lmax+1)^2
#define NELEM  5
#define INV_SQRT_C 0.08838834764831845f  // 1/sqrt(128)

// flattened duvw row counts of the U tensors
#define R3_0 729     // 1*9*9*9
#define R3_1 2187    // 3*9*9*9
#define R2_0 81
#define R2_1 243
#define R1_0 9
#define R1_1 27

#define MT 5         // M-tiles (of 16 nodes) per wave: 125 * 5 * 16 = 10000

// ---------------------------------------------------------------------------
// Kernel 1: fold element weights into the CG tensors:
//   UW[e, r, c] = sum_k U[r, k] * w[e, k, c]      (r = flattened d,u,v,w)
// One block (128 threads = channels) per (e, r).
// ---------------------------------------------------------------------------
__global__ void __launch_bounds__(CH)
uw_precompute(const float* __restrict__ U, const float* __restrict__ w,
              float* __restrict__ UW, int R, int K) {
    int br = blockIdx.x;
    int e  = br / R;
    int r  = br % R;
    int c  = threadIdx.x;
    const float* urow = U + (size_t)r * K;
    const float* wrow = w + ((size_t)e * K) * CH + c;
    float s = 0.f;
    for (int k = 0; k < K; ++k)
        s = fmaf(urow[k], wrow[(size_t)k * CH], s);
    UW[((size_t)e * R + r) * CH + c] = s;
}

// ---------------------------------------------------------------------------
// Kernel 2: symmetric contraction (correlation=3) per node, per channel.
// Block = 128 threads (one channel each), grid = N nodes.
// UW tables are c-fastest -> fully coalesced; ~8.4MB total so L2-resident.
// ---------------------------------------------------------------------------
__global__ void __launch_bounds__(CH)
symm_contract(const float* __restrict__ x,      // [N, C, L]
              const float* __restrict__ attrs,  // [N, E] one-hot
              const float* __restrict__ UW3_0, const float* __restrict__ UW2_0,
              const float* __restrict__ UW1_0,
              const float* __restrict__ UW3_1, const float* __restrict__ UW2_1,
              const float* __restrict__ UW1_1,
              float* __restrict__ out0,         // [N, C]
              float* __restrict__ out1) {       // [N, C, 3]
    __shared__ float xs[CH * LSQ];
    const int b = blockIdx.x;
    const int c = threadIdx.x;

    // stage this node's features through LDS (coalesced global reads)
    const float* xb = x + (size_t)b * CH * LSQ;
    for (int i = c; i < CH * LSQ; i += CH) xs[i] = xb[i];
    __syncthreads();

    float xl[LSQ];
#pragma unroll
    for (int l = 0; l < LSQ; ++l) xl[l] = xs[c * LSQ + l];  // stride 9: bank-conflict free

    // element id from one-hot row
    int e = 0;
#pragma unroll
    for (int j = 0; j < NELEM; ++j)
        if (attrs[(size_t)b * NELEM + j] > 0.5f) e = j;

    // ---- l_out = 0 (d = 1) ----
    {
        const float* P3 = UW3_0 + ((size_t)e * R3_0) * CH + c;
        const float* P2 = UW2_0 + ((size_t)e * R2_0) * CH + c;
        const float* P1 = UW1_0 + ((size_t)e * R1_0) * CH + c;
        float o = 0.f;
        for (int u = 0; u < LSQ; ++u) {
            float au = 0.f;
            for (int v = 0; v < LSQ; ++v) {
                float t = P2[(size_t)(u * LSQ + v) * CH];
#pragma unroll
                for (int w = 0; w < LSQ; ++w)
                    t = fmaf(P3[(size_t)((u * LSQ + v) * LSQ + w) * CH], xl[w], t);
                au = fmaf(t, xl[v], au);
            }
            o = fmaf(P1[(size_t)u * CH] + au, xl[u], o);
        }
        out0[(size_t)b * CH + c] = o;
    }

    // ---- l_out = 1 (d = 3) ----
    for (int d = 0; d < 3; ++d) {
        const float* P3 = UW3_1 + ((size_t)e * R3_1 + (size_t)d * 729) * CH + c;
        const float* P2 = UW2_1 + ((size_t)e * R2_1 + (size_t)d * 81) * CH + c;
        const float* P1 = UW1_1 + ((size_t)e * R1_1 + (size_t)d * 9) * CH + c;
        float o = 0.f;
        for (int u = 0; u < LSQ; ++u) {
            float au = 0.f;
            for (int v = 0; v < LSQ; ++v) {
                float t = P2[(size_t)(u * LSQ + v) * CH];
#pragma unroll
                for (int w = 0; w < LSQ; ++w)
                    t = fmaf(P3[(size_t)((u * LSQ + v) * LSQ + w) * CH], xl[w], t);
                au = fmaf(t, xl[v], au);
            }
            o = fmaf(P1[(size_t)u * CH] + au, xl[u], o);
        }
        out1[((size_t)b * CH + c) * 3 + d] = o;
    }
}

// ---------------------------------------------------------------------------
// Kernel 3: o3.Linear as fp32 WMMA (V_WMMA_F32_16X16X4_F32) + residual.
//   sec 0      : out[:,   j          ] = (out0 @ lin_w0) * s + sc
//   sec 1..3   : out[:, 128 + j*3 + d] = (out1[:,:,d] @ lin_w1) * s + sc
// Each wave computes MT=5 tiles of 16(node) x 16(j); K-loop = 128 in steps
// of 4. B fragment is loaded once per K-step and reused across the 5 WMMAs
// (independent accumulators -> no RAW hazard stalls, good pipelining).
// A 16x4 layout (ISA 7.12.2): lanes 0-15 hold K={0,1}, lanes 16-31 K={2,3}.
// C/D layout: VGPR r -> row (r + 8*half), col = lane&15.
// Branch-free A addressing: element (row,k) lives at src[(row*CH+k)*astride].
// ---------------------------------------------------------------------------
__global__ void __launch_bounds__(256)
linear_wmma(const float* __restrict__ out0,    // [N, C]
            const float* __restrict__ out1,    // [N, C, 3]
            const float* __restrict__ lin_w0,  // [C, C]
            const float* __restrict__ lin_w1,  // [C, C]
            const float* __restrict__ sc,      // [N, 512]
            float* __restrict__ out) {         // [N, 512]
    const int lane = threadIdx.x & 31;
    const int wave = threadIdx.x >> 5;
    const int half = lane >> 4;   // 0: lanes 0-15, 1: lanes 16-31
    const int n    = lane & 15;   // output column within tile (B/C/D)
    const int m    = lane & 15;   // A row within tile
    const int b0   = blockIdx.x * (16 * MT);   // 125 blocks * 80 nodes = 10000
    const int j0   = wave * 16;                // 8 waves cover 128 output channels
    const int sec  = blockIdx.y;               // 0 = l0 ; 1..3 = l1 components

    const float*  W       = (sec == 0) ? lin_w0 : lin_w1;
    const float*  src     = (sec == 0) ? out0 : (out1 + (sec - 1));
    const size_t  astride = (sec == 0) ? 1 : 3;

    v8f acc[MT];
#pragma unroll
    for (int t = 0; t < MT; ++t)
        acc[t] = (v8f){0.f, 0.f, 0.f, 0.f, 0.f, 0.f, 0.f, 0.f};

    for (int k0 = 0; k0 < CH; k0 += 4) {
        const int ka = k0 + 2 * half;          // this half-wave's K pair
        v2f bf;
        bf.x = W[(size_t)ka * CH + j0 + n];
        bf.y = W[(size_t)(ka + 1) * CH + j0 + n];
#pragma unroll
        for (int t = 0; t < MT; ++t) {
            const int row = b0 + t * 16 + m;
            const float* p = src + ((size_t)row * CH + ka) * astride;
            v2f a;
            a.x = p[0];
            a.y = p[astride];
            acc[t] = __builtin_amdgcn_wmma_f32_16x16x4_f32(
                /*neg_a=*/false, a, /*neg_b=*/false, bf,
                /*c_mod=*/(short)0, acc[t], /*reuse_a=*/false, /*reuse_b=*/false);
        }
    }

#pragma unroll
    for (int t = 0; t < MT; ++t) {
#pragma unroll
        for (int r = 0; r < 8; ++r) {
            const int brow = b0 + t * 16 + r + 8 * half;
            const int col  = (sec == 0) ? (j0 + n) : (CH + (j0 + n) * 3 + (sec - 1));
            const size_t idx = (size_t)brow * 512 + col;
            out[idx] = acc[t][r] * INV_SQRT_C + sc[idx];
        }
    }
}

// ---------------------------------------------------------------------------
extern "C" void kernel_launch(void* const* d_in, const int* in_sizes, int n_in,
                              void* d_out, int out_size, void* d_ws, size_t ws_size,
                              hipStream_t stream) {
    const float* node_feats = (const float*)d_in[0];
    const float* node_attrs = (const float*)d_in[1];
    const float* sc         = (const float*)d_in[2];
    const float* U3_0 = (const float*)d_in[3];
    const float* U2_0 = (const float*)d_in[4];
    const float* U1_0 = (const float*)d_in[5];
    const float* w3_0 = (const float*)d_in[6];
    const float* w2_0 = (const float*)d_in[7];
    const float* w1_0 = (const float*)d_in[8];
    const float* U3_1 = (const float*)d_in[9];
    const float* U2_1 = (const float*)d_in[10];
    const float* U1_1 = (const float*)d_in[11];
    const float* w3_1 = (const float*)d_in[12];
    const float* w2_1 = (const float*)d_in[13];
    const float* w1_1 = (const float*)d_in[14];
    const float* lin_w0 = (const float*)d_in[15];
    const float* lin_w1 = (const float*)d_in[16];
    float* out = (float*)d_out;

    // workspace layout (floats)
    float* ws    = (float*)d_ws;
    float* UW3_0 = ws;                                    // 5*729*128   = 466560
    float* UW3_1 = UW3_0 + (size_t)NELEM * R3_0 * CH;     // 5*2187*128  = 1399680
    float* UW2_0 = UW3_1 + (size_t)NELEM * R3_1 * CH;     // 5*81*128    = 51840
    float* UW2_1 = UW2_0 + (size_t)NELEM * R2_0 * CH;     // 5*243*128   = 155520
    float* UW1_0 = UW2_1 + (size_t)NELEM * R2_1 * CH;     // 5*9*128     = 5760
    float* UW1_1 = UW1_0 + (size_t)NELEM * R1_0 * CH;     // 5*27*128    = 17280
    float* o0    = UW1_1 + (size_t)NELEM * R1_1 * CH;     // 10000*128   = 1280000
    float* o1    = o0 + (size_t)NNODES * CH;              // 10000*384   = 3840000

    // 1) fold element weights into CG tensors (tiny)
    uw_precompute<<<NELEM * R3_0, CH, 0, stream>>>(U3_0, w3_0, UW3_0, R3_0, 10);
    uw_precompute<<<NELEM * R3_1, CH, 0, stream>>>(U3_1, w3_1, UW3_1, R3_1, 12);
    uw_precompute<<<NELEM * R2_0, CH, 0, stream>>>(U2_0, w2_0, UW2_0, R2_0, 3);
    uw_precompute<<<NELEM * R2_1, CH, 0, stream>>>(U2_1, w2_1, UW2_1, R2_1, 4);
    uw_precompute<<<NELEM * R1_0, CH, 0, stream>>>(U1_0, w1_0, UW1_0, R1_0, 1);
    uw_precompute<<<NELEM * R1_1, CH, 0, stream>>>(U1_1, w1_1, UW1_1, R1_1, 1);

    // 2) symmetric contraction (VALU fp32, UW tables L2-resident)
    symm_contract<<<NNODES, CH, 0, stream>>>(node_feats, node_attrs,
                                             UW3_0, UW2_0, UW1_0,
                                             UW3_1, UW2_1, UW1_1,
                                             o0, o1);

    // 3) o3.Linear via fp32 WMMA + residual (writes every output element)
    linear_wmma<<<dim3(NNODES / (16 * MT), 4), 256, 0, stream>>>(o0, o1,
                                                                 lin_w0, lin_w1,
                                                                 sc, out);
}